// MultiHeadAttention_52733608460565
// MI455X (gfx1250) — compile-verified
//
#include <hip/hip_runtime.h>
#include <hip/hip_bf16.h>
#include <math.h>

#define N_NODES   4096
#define HIDDEN    512
#define NUM_HEADS 8
#define HEAD_DIM  64
#define NUM_GRAPHS 16

typedef __attribute__((ext_vector_type(2))) float v2f;
typedef __attribute__((ext_vector_type(8))) float v8f;

__device__ __forceinline__ v8f wmma_f32(v2f a, v2f b, v8f c) {
  // (neg_a, A, neg_b, B, c_mod, C, reuse_a, reuse_b) -> v_wmma_f32_16x16x4_f32
  return __builtin_amdgcn_wmma_f32_16x16x4_f32(false, a, false, b, (short)0, c,
                                               false, false);
}

// --------------------------------------------------------------------------
// Kernel 1: per-graph start offsets. batch is sorted ascending in [0,16).
// starts[g] = first index with batch[idx] >= g, starts[16] = N_NODES.
// --------------------------------------------------------------------------
__global__ void graph_starts_kernel(const int* __restrict__ batch,
                                    int* __restrict__ starts) {
  int t = threadIdx.x;
  if (t > NUM_GRAPHS) return;
  int lo = 0, hi = N_NODES;
  while (lo < hi) {
    int mid = (lo + hi) >> 1;
    if (batch[mid] < t) lo = mid + 1; else hi = mid;
  }
  starts[t] = lo;
}

// --------------------------------------------------------------------------
// Kernel 2/4: Y = X @ W^T + b   (X:[4096,512], W:[512,512] row-major, both
// K-contiguous -> A and B fragments load with the same pattern).
// Register-blocked 2x2: each wave owns a 32x32 output tile, 4 accumulators,
// 2 A-frags + 2 B-frags per K-step feeding 4 WMMAs (1:1 wmma:load ratio).
// blockIdx.y selects among up to 3 (W,b,Y) triples (fused QKV).
// --------------------------------------------------------------------------
__global__ __launch_bounds__(256) void proj_kernel(
    const float* __restrict__ X,
    const float* __restrict__ W0, const float* __restrict__ B0, float* __restrict__ Y0,
    const float* __restrict__ W1, const float* __restrict__ B1, float* __restrict__ Y1,
    const float* __restrict__ W2, const float* __restrict__ B2, float* __restrict__ Y2) {
  const float* W; const float* B; float* Y;
  if (blockIdx.y == 0)      { W = W0; B = B0; Y = Y0; }
  else if (blockIdx.y == 1) { W = W1; B = B1; Y = Y1; }
  else                      { W = W2; B = B2; Y = Y2; }

  const int lane = threadIdx.x & 31;
  const int wave = threadIdx.x >> 5;
  const int gw   = blockIdx.x * 8 + wave;        // global wave id
  const int tilesN = HIDDEN / 32;                // 16 (32-wide N tiles)
  const int mt = gw / tilesN;                    // 0..127 (32-tall M tiles)
  const int nt = gw % tilesN;

  const int half = lane >> 4;
  const int ln   = lane & 15;
  const int koff = half * 2;                     // K sub-offset per half-wave

  // A frags: lanes 0-15 -> M=lane, K=k+0/k+1; lanes 16-31 -> K=k+2/k+3.
  const float* xrow0 = X + (size_t)(mt * 32 + ln) * HIDDEN + koff;
  const float* xrow1 = xrow0 + (size_t)16 * HIDDEN;
  // B frags (B = W^T): b = W[col][k+koff..+1], same pattern as A.
  const float* wrow0 = W + (size_t)(nt * 32 + ln) * HIDDEN + koff;
  const float* wrow1 = wrow0 + (size_t)16 * HIDDEN;

  v8f acc00 = {}, acc01 = {}, acc10 = {}, acc11 = {};
#pragma unroll 4
  for (int k = 0; k < HIDDEN; k += 4) {
    v2f a0 = *(const v2f*)(xrow0 + k);
    v2f a1 = *(const v2f*)(xrow1 + k);
    v2f b0 = *(const v2f*)(wrow0 + k);
    v2f b1 = *(const v2f*)(wrow1 + k);
    acc00 = wmma_f32(a0, b0, acc00);
    acc01 = wmma_f32(a0, b1, acc01);
    acc10 = wmma_f32(a1, b0, acc10);
    acc11 = wmma_f32(a1, b1, acc11);
  }

  const int col0 = nt * 32 + ln;
  const int col1 = col0 + 16;
  const float bi0 = B[col0];
  const float bi1 = B[col1];
#pragma unroll
  for (int i = 0; i < 8; ++i) {
    const int row0 = mt * 32 + i + half * 8;     // C/D layout
    const int row1 = row0 + 16;
    Y[(size_t)row0 * HIDDEN + col0] = acc00[i] + bi0;
    Y[(size_t)row0 * HIDDEN + col1] = acc01[i] + bi1;
    Y[(size_t)row1 * HIDDEN + col0] = acc10[i] + bi0;
    Y[(size_t)row1 * HIDDEN + col1] = acc11[i] + bi1;
  }
}

// --------------------------------------------------------------------------
// Kernel 3: flash attention per (head, 16-row Q tile), one wave each.
// KV sweep restricted to the contiguous graph range of this Q tile.
// --------------------------------------------------------------------------
__global__ __launch_bounds__(256) void attn_kernel(
    const float* __restrict__ Q, const float* __restrict__ K,
    const float* __restrict__ V, const int* __restrict__ batch,
    const int* __restrict__ starts, float* __restrict__ O) {
  __shared__ float pstage[8][256];               // per-wave 16x16 P staging

  const int lane = threadIdx.x & 31;
  const int wave = threadIdx.x >> 5;
  const int gw   = blockIdx.x * 8 + wave;
  const int qtiles = N_NODES / 16;               // 256
  const int h  = gw / qtiles;
  const int qt = gw % qtiles;

  const int half  = lane >> 4;
  const int ln    = lane & 15;
  const int koff  = half * 2;
  const int hbase = h * HEAD_DIM;
  const int qbase = qt * 16;

  // Preload Q A-fragments for all 16 K-steps (head_dim 64 / 4).
  v2f qf[16];
  const float* qrow = Q + (size_t)(qbase + ln) * HIDDEN + hbase + koff;
#pragma unroll
  for (int ks = 0; ks < 16; ++ks) qf[ks] = *(const v2f*)(qrow + 4 * ks);

  // Graph ids of the 8 rows this lane accumulates (rows i + half*8).
  int qb[8];
#pragma unroll
  for (int i = 0; i < 8; ++i) qb[i] = batch[qbase + i + half * 8];

  float m[8], l[8];
#pragma unroll
  for (int i = 0; i < 8; ++i) { m[i] = -__builtin_inff(); l[i] = 0.f; }
  v8f o[4] = {};                                  // 16x64 O accumulator

  const int g_lo = batch[qbase];
  const int g_hi = batch[qbase + 15];
  const int kv_start = starts[g_lo] & ~15;
  int kv_end = (starts[g_hi + 1] + 15) & ~15;
  if (kv_end > N_NODES) kv_end = N_NODES;

  const float scale = 0.125f;                     // 1/sqrt(64)

  for (int ct = kv_start; ct < kv_end; ct += 16) {
    // ---- S tile = Q . K^T : 16 WMMAs over K=64 ----
    v8f s = {};
    const float* krow = K + (size_t)(ct + ln) * HIDDEN + hbase + koff;
#pragma unroll
    for (int ks = 0; ks < 16; ++ks) {
      v2f kb = *(const v2f*)(krow + 4 * ks);
      s = wmma_f32(qf[ks], kb, s);
    }

    const int cb = batch[ct + ln];                // this lane's column graph
#pragma unroll
    for (int i = 0; i < 8; ++i) {
      float sv = s[i] * scale;
      if (cb != qb[i]) sv = -__builtin_inff();    // block-diagonal mask
      // row max across the 16 lanes holding this row (stays in half-wave)
      float tm = sv;
      tm = fmaxf(tm, __shfl_xor(tm, 1));
      tm = fmaxf(tm, __shfl_xor(tm, 2));
      tm = fmaxf(tm, __shfl_xor(tm, 4));
      tm = fmaxf(tm, __shfl_xor(tm, 8));
      const float mn = fmaxf(m[i], tm);
      float corr, p;
      if (mn == -__builtin_inff()) { corr = 1.f; p = 0.f; }
      else { corr = __expf(m[i] - mn); p = __expf(sv - mn); }
      float ps = p;
      ps += __shfl_xor(ps, 1);
      ps += __shfl_xor(ps, 2);
      ps += __shfl_xor(ps, 4);
      ps += __shfl_xor(ps, 8);
      l[i] = l[i] * corr + ps;
      m[i] = mn;
      o[0][i] *= corr; o[1][i] *= corr; o[2][i] *= corr; o[3][i] *= corr;
      pstage[wave][(i + half * 8) * 16 + ln] = p; // C/D layout -> row-major
    }

    // P A-fragments from LDS (per-wave private; DS ops in-order per wave)
    v2f pf[4];
#pragma unroll
    for (int ks = 0; ks < 4; ++ks) {
      pf[ks][0] = pstage[wave][ln * 16 + 4 * ks + koff];
      pf[ks][1] = pstage[wave][ln * 16 + 4 * ks + koff + 1];
    }

    // ---- O += P . V : 16 WMMAs (4 N-slices x K=16) ----
#pragma unroll
    for (int nn = 0; nn < 4; ++nn) {
#pragma unroll
      for (int ks = 0; ks < 4; ++ks) {
        const size_t vr = (size_t)(ct + 4 * ks + koff) * HIDDEN + hbase + 16 * nn + ln;
        v2f vb;
        vb[0] = V[vr];
        vb[1] = V[vr + HIDDEN];
        o[nn] = wmma_f32(pf[ks], vb, o[nn]);
      }
    }
  }

  // normalize and store (every row attends at least to itself -> l > 0)
  float invl[8];
#pragma unroll
  for (int i = 0; i < 8; ++i) invl[i] = 1.f / l[i];
#pragma unroll
  for (int nn = 0; nn < 4; ++nn)
#pragma unroll
    for (int i = 0; i < 8; ++i) {
      const int row = qbase + i + half * 8;
      O[(size_t)row * HIDDEN + hbase + 16 * nn + ln] = o[nn][i] * invl[i];
    }
}

// --------------------------------------------------------------------------
extern "C" void kernel_launch(void* const* d_in, const int* in_sizes, int n_in,
                              void* d_out, int out_size, void* d_ws, size_t ws_size,
                              hipStream_t stream) {
  const float* x     = (const float*)d_in[0];
  const int*   batch = (const int*)d_in[1];
  const float* Wq = (const float*)d_in[2];
  const float* bq = (const float*)d_in[3];
  const float* Wk = (const float*)d_in[4];
  const float* bk = (const float*)d_in[5];
  const float* Wv = (const float*)d_in[6];
  const float* bv = (const float*)d_in[7];
  const float* Wo = (const float*)d_in[8];
  const float* bo = (const float*)d_in[9];
  float* out = (float*)d_out;

  const size_t mat = (size_t)N_NODES * HIDDEN;
  float* Qb = (float*)d_ws;
  float* Kb = Qb + mat;
  float* Vb = Kb + mat;
  float* Ab = Vb + mat;
  int* starts = (int*)(Ab + mat);

  graph_starts_kernel<<<1, 32, 0, stream>>>(batch, starts);

  // 32x32 tiles: (4096/32)*(512/32) = 2048 waves per matrix, 8 waves/block
  const int tiles = (N_NODES / 32) * (HIDDEN / 32);      // 2048
  dim3 pgrid(tiles / 8, 3, 1);
  proj_kernel<<<pgrid, 256, 0, stream>>>(x, Wq, bq, Qb, Wk, bk, Kb, Wv, bv, Vb);

  const int awaves = NUM_HEADS * (N_NODES / 16);          // 2048 waves
  attn_kernel<<<awaves / 8, 256, 0, stream>>>(Qb, Kb, Vb, batch, starts, Ab);

  dim3 ogrid(tiles / 8, 1, 1);
  proj_kernel<<<ogrid, 256, 0, stream>>>(Ab, Wo, bo, out, Wo, bo, out, Wo, bo, out);
}